// UniformShardedEmbeddingBags_16149077033312
// MI455X (gfx1250) — compile-verified
//
#include <hip/hip_runtime.h>
#include <hip/hip_bf16.h>

// Problem dims (match reference)
#define E_DIM 200000
#define T_DIM 16
#define D_DIM 64
#define B_DIM 4096
#define L_DIM 20

#define ROW_BYTES (D_DIM * 4)              // 256 B: one (e,t) row slice, contiguous
#define PAIR_BYTES (L_DIM * ROW_BYTES)     // 5120 B staged per bag
#define BUF_BYTES (2 * PAIR_BYTES)         // double buffer per wave (10 KB)
#define WAVES_PER_BLOCK 8
#define PAIRS_PER_WAVE 4
#define TOTAL_PAIRS (B_DIM * T_DIM)        // 65536
#define NBLOCKS (TOTAL_PAIRS / (WAVES_PER_BLOCK * PAIRS_PER_WAVE)) // 2048

// ---- CDNA5 async global->LDS path (guarded so compile never breaks) ----
#if __has_builtin(__builtin_amdgcn_global_load_async_to_lds_b128)
#define HAVE_ASYNC 1
#else
#define HAVE_ASYNC 0
#endif

// Builtin param types (from hipcc diagnostic): global v4i32* and LDS v4i32*.
typedef int v4i __attribute__((ext_vector_type(4)));
typedef __attribute__((address_space(1))) v4i gv4i;  // global (printed as __device__)
typedef __attribute__((address_space(3))) v4i lv4i;  // LDS

#if HAVE_ASYNC
#if __has_builtin(__builtin_amdgcn_s_wait_asynccnt)
#define WAIT_ASYNCCNT(n) __builtin_amdgcn_s_wait_asynccnt(n)
#else
#define WAIT_ASYNCCNT(n) asm volatile("s_wait_asynccnt %0" ::"n"(n) : "memory")
#endif
// ASYNC writes to LDS are unordered vs DS reads from the same wave: drain DScnt
// before re-targeting a buffer that was just read.
#define WAIT_DSCNT() asm volatile("s_wait_dscnt 0x0" ::: "memory")
#else
#define WAIT_ASYNCCNT(n) ((void)0)
#define WAIT_DSCNT() ((void)0)
#endif

// Issue the 20 row-gathers of one bag into an LDS buffer.
// bagIdxLane: lane l (<20) holds embedding index for bag slot l (wave-uniform data).
// One b128 per lane; lanes 0-15 cover row 2j, lanes 16-31 cover row 2j+1.
__device__ __forceinline__ void issue_bag_loads(const char* __restrict__ wbase_t,
                                                int bagIdxLane, char* buf, int lane) {
#pragma unroll
  for (int j = 0; j < L_DIM / 2; ++j) {
    const int r = 2 * j + (lane >> 4);
    const unsigned e = (unsigned)__shfl(bagIdxLane, r, 32);
    // byte offset e*4096 fits in 32 bits (table is 819 MB)
    const char* g = wbase_t + (size_t)e * (size_t)(T_DIM * D_DIM * 4) +
                    (size_t)((lane & 15) * 16);
    char* l = buf + j * 512 + lane * 16;  // == buf + row*256 + (lane&15)*16
#if HAVE_ASYNC
    __builtin_amdgcn_global_load_async_to_lds_b128((gv4i*)(void*)g, (lv4i*)l, 0, 0);
#else
    *(float4*)l = *(const float4*)g;  // synchronous fallback through VGPRs
#endif
  }
}

// Reduce one staged bag: each lane owns 2 of the 64 output floats.
__device__ __forceinline__ float2 reduce_bag(const char* buf, int lane) {
  float2 acc = make_float2(0.f, 0.f);
#pragma unroll
  for (int r = 0; r < L_DIM; ++r) {
    const float2 v = *(const float2*)(buf + r * ROW_BYTES + lane * 8);
    acc.x += v.x;
    acc.y += v.y;
  }
  return acc;
}

__global__ __launch_bounds__(WAVES_PER_BLOCK * 32, 1) void
UniformShardedEmbeddingBags_kernel(const float* __restrict__ w,
                                   const int* __restrict__ idx,
                                   float* __restrict__ out) {
  extern __shared__ char smem[];  // WAVES_PER_BLOCK * BUF_BYTES (80 KB)
  const int lane = threadIdx.x & 31;
  const int wave = threadIdx.x >> 5;
  char* mybuf = smem + wave * BUF_BYTES;

  const int waveGlobal = blockIdx.x * WAVES_PER_BLOCK + wave;
  const int P0 = waveGlobal * PAIRS_PER_WAVE;  // first (b,t) pair for this wave
  const char* wbytes = (const char*)w;
  char* outbytes = (char*)out;

  // Prologue: stage bag P0 into buffer 0.
  {
    const int t0 = P0 & (T_DIM - 1);
    const int id0 = (lane < L_DIM) ? idx[P0 * L_DIM + lane] : 0;
    issue_bag_loads(wbytes + t0 * ROW_BYTES, id0, mybuf, lane);
  }

#pragma unroll
  for (int pp = 0; pp < PAIRS_PER_WAVE; ++pp) {
    if (pp + 1 < PAIRS_PER_WAVE) {
      // Stage next bag into the other buffer, then wait for the current one.
      const int Pn = P0 + pp + 1;
      const int tn = Pn & (T_DIM - 1);
      const int idn = (lane < L_DIM) ? idx[Pn * L_DIM + lane] : 0;
      WAIT_DSCNT();  // prior reads of the buffer being overwritten are done
      issue_bag_loads(wbytes + tn * ROW_BYTES, idn,
                      mybuf + ((pp + 1) & 1) * PAIR_BYTES, lane);
      WAIT_ASYNCCNT(L_DIM / 2);  // 10 newer ops may remain outstanding
    } else {
      WAIT_ASYNCCNT(0);
    }
    const float2 acc = reduce_bag(mybuf + (pp & 1) * PAIR_BYTES, lane);
    // Coalesced 256 B store: out[P*64 + lane*2 .. +1]
    *((float2*)(outbytes + (size_t)(P0 + pp) * ROW_BYTES) + lane) = acc;
  }
}

extern "C" void kernel_launch(void* const* d_in, const int* in_sizes, int n_in,
                              void* d_out, int out_size, void* d_ws, size_t ws_size,
                              hipStream_t stream) {
  const float* w = (const float*)d_in[0];   // weights [E, T, D] fp32
  const int* idx = (const int*)d_in[1];     // indices [B, T, L] int
  float* out = (float*)d_out;               // [B, T, D] fp32
  (void)in_sizes; (void)n_in; (void)out_size; (void)d_ws; (void)ws_size;

  dim3 grid(NBLOCKS);
  dim3 block(WAVES_PER_BLOCK * 32);
  const size_t lds_bytes = (size_t)WAVES_PER_BLOCK * BUF_BYTES;  // 81920 B
  hipLaunchKernelGGL(UniformShardedEmbeddingBags_kernel, grid, block, lds_bytes,
                     stream, w, idx, out);
}